// FeatureEncoder_41515153883692
// MI455X (gfx1250) — compile-verified
//
#include <hip/hip_runtime.h>

// ---------------------------------------------------------------------------
// FeatureEncoder for MI455X (gfx1250): per-point MLP via V_WMMA_F32_16X16X4_F32
// with 4x M-tile B-fragment reuse + dense-key unique/prefix-scan + encoded
// atomicMax segment pooling.
// ---------------------------------------------------------------------------

#define GXD 64
#define GYD 64
#define GZD 32
#define BATCH 4
#define NKEYS (BATCH * GXD * GYD * GZD)   // 524288
#define EPSV 1e-5f

typedef __attribute__((ext_vector_type(2))) float v2f;
typedef __attribute__((ext_vector_type(8))) float v8f;

// ---- workspace layout (32-bit words) ----
#define OFF_FLAGS 0
#define OFF_SCAN  (NKEYS)
#define OFF_BSUM  (2 * NKEYS)
#define OFF_M     (2 * NKEYS + 512)
#define OFF_P     (2 * NKEYS + 516)
// param block sub-offsets (relative to OFF_P, float words)
#define P_SIN 0        // 12  input-BN scale (9 + pad)
#define P_TIN 12       // 12  input-BN shift
#define P_SC0 24       // 64
#define P_SH0 88       // 64
#define P_SC1 152      // 128
#define P_SH1 280      // 128
#define P_SC2 408      // 256
#define P_SH2 664      // 256
#define P_SC3 920      // 64  (==1.0, last layer has no BN)
#define P_SH3 984      // 64  (==bias3)
// fragment-packed weights: pairs of consecutive K rows interleaved per column
#define P_WP0 1048               // 12*64   = 768
#define P_WP1 (P_WP0 + 768)      // 64*128  = 8192
#define P_WP2 (P_WP1 + 8192)     // 128*256 = 32768
#define P_WP3 (P_WP2 + 32768)    // 256*64  = 16384
#define P_END (P_WP3 + 16384)

// Order-preserving float<->uint encoding for atomicMax (handles negatives).
__device__ __forceinline__ unsigned encF(float f) {
  unsigned u = __float_as_uint(f);
  return (u & 0x80000000u) ? ~u : (u | 0x80000000u);
}
__device__ __forceinline__ float decF(unsigned e) {
  unsigned u = (e & 0x80000000u) ? (e & 0x7FFFFFFFu) : ~e;
  return __uint_as_float(u);
}
#define ENC_NEG_INF 0x007FFFFFu   // encF(-inf)

// ---------------------------------------------------------------------------
__global__ void k_fill_u32(unsigned* p, unsigned val, int n) {
  int i = blockIdx.x * blockDim.x + threadIdx.x;
  if (i < n) p[i] = val;
}

// Fold BN into per-channel scale/shift.
__global__ void k_param_prep(
    const float* g0, const float* be0, const float* m0, const float* v0,
    const float* g1, const float* be1, const float* m1, const float* v1,
    const float* g2, const float* be2, const float* m2, const float* v2,
    const float* g3, const float* be3, const float* m3, const float* v3,
    const float* b0, const float* b1, const float* b2, const float* b3,
    float* wsP) {
  int tid = threadIdx.x;
  for (int c = tid; c < 12; c += blockDim.x) {
    if (c < 9) {
      float s = g0[c] * rsqrtf(v0[c] + EPSV);
      wsP[P_SIN + c] = s;
      wsP[P_TIN + c] = be0[c] - m0[c] * s;
    } else {
      wsP[P_SIN + c] = 0.f;
      wsP[P_TIN + c] = 0.f;
    }
  }
  for (int c = tid; c < 64; c += blockDim.x) {
    float s = g1[c] * rsqrtf(v1[c] + EPSV);
    wsP[P_SC0 + c] = s;
    wsP[P_SH0 + c] = b0[c] * s + be1[c] - m1[c] * s;
  }
  for (int c = tid; c < 128; c += blockDim.x) {
    float s = g2[c] * rsqrtf(v2[c] + EPSV);
    wsP[P_SC1 + c] = s;
    wsP[P_SH1 + c] = b1[c] * s + be2[c] - m2[c] * s;
  }
  for (int c = tid; c < 256; c += blockDim.x) {
    float s = g3[c] * rsqrtf(v3[c] + EPSV);
    wsP[P_SC2 + c] = s;
    wsP[P_SH2 + c] = b2[c] * s + be3[c] - m3[c] * s;
  }
  for (int c = tid; c < 64; c += blockDim.x) {
    wsP[P_SC3 + c] = 1.f;
    wsP[P_SH3 + c] = b3[c];
  }
}

// Pack weights (srcRows x C, row-major, zero-pad K) into B-fragment order:
// dst[(pair*C + col)*2 + j] = W[(2*pair + j)*C + col]  -> one b64 per lane/frag.
__global__ void k_pack(const float* __restrict__ W, float* __restrict__ dst,
                       int K, int C, int srcRows) {
  int d = blockIdx.x * blockDim.x + threadIdx.x;
  if (d >= K * C) return;
  int pair = d / (2 * C);
  int rem = d % (2 * C);
  int col = rem >> 1;
  int j = rem & 1;
  int r = pair * 2 + j;
  dst[d] = (r < srcRows) ? W[r * C + col] : 0.f;
}

__global__ void k_mark(const int* __restrict__ pidx, unsigned* __restrict__ flags,
                       int N, int P) {
  int p = blockIdx.x * blockDim.x + threadIdx.x;
  if (p >= N) return;
  int b = p / P;
  int q = p * 3;
  int key = ((b * GXD + pidx[q]) * GYD + pidx[q + 1]) * GZD + pidx[q + 2];
  flags[key] = 1u;
}

// 512 blocks x 256 threads; 1024 flags per block -> block sums
__global__ void k_bsum(const unsigned* __restrict__ flags, unsigned* __restrict__ bsum) {
  __shared__ unsigned sm[256];
  int t = threadIdx.x;
  int base = blockIdx.x * 1024 + t * 4;
  unsigned s = flags[base] + flags[base + 1] + flags[base + 2] + flags[base + 3];
  sm[t] = s;
  __syncthreads();
  for (int off = 128; off > 0; off >>= 1) {
    if (t < off) sm[t] += sm[t + off];
    __syncthreads();
  }
  if (t == 0) bsum[blockIdx.x] = sm[0];
}

// exclusive scan of 512 block sums, write total unique count M
__global__ void k_bscan(unsigned* bsum, unsigned* Mout) {
  __shared__ unsigned sm[512];
  int t = threadIdx.x;
  unsigned v = bsum[t];
  sm[t] = v;
  __syncthreads();
  for (int off = 1; off < 512; off <<= 1) {
    unsigned a = (t >= off) ? sm[t - off] : 0u;
    __syncthreads();
    sm[t] += a;
    __syncthreads();
  }
  bsum[t] = sm[t] - v;   // exclusive
  if (t == 511) Mout[0] = sm[511];
}

// final exclusive scan over all keys
__global__ void k_scan(const unsigned* __restrict__ flags,
                       const unsigned* __restrict__ bsum,
                       unsigned* __restrict__ scan) {
  __shared__ unsigned sm[256];
  int t = threadIdx.x;
  int base = blockIdx.x * 1024 + t * 4;
  unsigned f0 = flags[base], f1 = flags[base + 1], f2 = flags[base + 2], f3 = flags[base + 3];
  unsigned s = f0 + f1 + f2 + f3;
  unsigned own = s;
  sm[t] = s;
  __syncthreads();
  for (int off = 1; off < 256; off <<= 1) {
    unsigned a = (t >= off) ? sm[t - off] : 0u;
    __syncthreads();
    sm[t] += a;
    __syncthreads();
  }
  unsigned ex = sm[t] - own + bsum[blockIdx.x];
  scan[base] = ex;
  scan[base + 1] = ex + f0;
  scan[base + 2] = ex + f0 + f1;
  scan[base + 3] = ex + f0 + f1 + f2;
}

// emit decoded (b,x,y,z) at rank position (ascending key == jnp.unique sort order)
__global__ void k_emit(const unsigned* __restrict__ flags,
                       const unsigned* __restrict__ scan, int* __restrict__ outIdx) {
  int k = blockIdx.x * blockDim.x + threadIdx.x;
  if (k >= NKEYS) return;
  if (!flags[k]) return;
  unsigned r = scan[k];
  int z = k % GZD;
  int rr = k / GZD;
  int y = rr % GYD;
  rr /= GYD;
  int x = rr % GXD;
  int b = rr / GXD;
  outIdx[r * 4 + 0] = b;
  outIdx[r * 4 + 1] = x;
  outIdx[r * 4 + 2] = y;
  outIdx[r * 4 + 3] = z;
}

// ---------------------------------------------------------------------------
// One GEMM layer on the f32 WMMA pipe. ROWS=64 points per wave (4 M-tiles):
// each B fragment (one global_load_b64 from packed weights) feeds 4 WMMAs on
// independent accumulators -> 32 FLOP per B-byte, deep pipeline.
// ---------------------------------------------------------------------------
template <int K, int C, bool LAST>
__device__ __forceinline__ void run_layer(
    const float* binLds, float* boutLds, const float* __restrict__ Wp,
    const float* __restrict__ sc, const float* __restrict__ sh,
    const int* invBuf, unsigned* __restrict__ pooled, long base, int N,
    int l16, int half) {
  constexpr int MT = 4;  // 64 rows
  for (int nt = 0; nt < C / 16; ++nt) {
    int col = nt * 16 + l16;
    v8f acc[MT];
#pragma unroll
    for (int m = 0; m < MT; ++m) acc[m] = (v8f){0.f, 0.f, 0.f, 0.f, 0.f, 0.f, 0.f, 0.f};
#pragma unroll 4
    for (int kt = 0; kt < K / 4; ++kt) {
      // B fragment: rows {kt*4+2*half, +1}, col = l16  (packed, contiguous)
      const float2 bb = *(const float2*)(Wp + ((kt * 2 + half) * C + col) * 2);
      v2f b;
      b[0] = bb.x;
      b[1] = bb.y;
#pragma unroll
      for (int m = 0; m < MT; ++m) {
        const float2 aa =
            *(const float2*)(binLds + (m * 16 + l16) * K + kt * 4 + half * 2);
        v2f a;
        a[0] = aa.x;
        a[1] = aa.y;
        acc[m] = __builtin_amdgcn_wmma_f32_16x16x4_f32(
            false, a, false, b, (short)0, acc[m], false, false);
      }
    }
    float s = sc[col];
    float t = sh[col];
#pragma unroll
    for (int m = 0; m < MT; ++m) {
#pragma unroll
      for (int i = 0; i < 8; ++i) {
        int row = m * 16 + i + half * 8;   // C/D layout: rows i / i+8
        if (!LAST) {
          boutLds[row * C + col] = fmaxf(acc[m][i] * s + t, 0.f);  // BN+ReLU
        } else {
          long p = base + row;
          if (p < (long)N) {
            float y = acc[m][i] * s + t;   // bias only
            atomicMax(&pooled[(long)invBuf[row] * 64 + col], encF(y));
          }
        }
      }
    }
  }
}

// ---------------------------------------------------------------------------
// Main MLP kernel: 1 wave per block, 64 points per wave, 96KB LDS overlay:
//   even layers: in @0, out @8192 floats; odd layers: in @8192, out @0.
//   L0 in 64x12(768)  -> out 64x64(4096)   | L1 in @8192 -> out 64x128(8192)@0
//   L2 in @0 (8192)   -> out 64x256(16384)@8192 | L3 in @8192 -> scatter
// ---------------------------------------------------------------------------
__global__ void __launch_bounds__(32) k_mlp(
    const float* __restrict__ feat, const int* __restrict__ pidx,
    const unsigned* __restrict__ scan, const float* __restrict__ wsP,
    unsigned* __restrict__ pooled, int N, int P) {
  __shared__ float smem[24576];   // 96 KB (of 320 KB/WGP)
  __shared__ int invBuf[64];

  int lane = threadIdx.x;
  int l16 = lane & 15;
  int half = lane >> 4;
  long base = (long)blockIdx.x * 64;

  // Warm L2/L1 with the largest weight block while we stage inputs.
  __builtin_prefetch((const void*)((const char*)(wsP + P_WP2) + lane * 4096), 0, 1);

  // Stage input tile: apply input BN, pad K 9->12, compute inverse index.
  for (int r0 = lane; r0 < 64; r0 += 32) {
    long p = base + r0;
    if (p < (long)N) {
#pragma unroll
      for (int k = 0; k < 9; ++k)
        smem[r0 * 12 + k] = feat[p * 9 + k] * wsP[P_SIN + k] + wsP[P_TIN + k];
      smem[r0 * 12 + 9] = 0.f;
      smem[r0 * 12 + 10] = 0.f;
      smem[r0 * 12 + 11] = 0.f;
      int b = (int)(p / P);
      long q = p * 3;
      int key = ((b * GXD + pidx[q]) * GYD + pidx[q + 1]) * GZD + pidx[q + 2];
      invBuf[r0] = (int)scan[key];
    } else {
#pragma unroll
      for (int k = 0; k < 12; ++k) smem[r0 * 12 + k] = 0.f;
      invBuf[r0] = 0;
    }
  }
  __syncthreads();

  run_layer<12, 64, false>(smem, smem + 8192, wsP + P_WP0, wsP + P_SC0,
                           wsP + P_SH0, invBuf, pooled, base, N, l16, half);
  __syncthreads();
  run_layer<64, 128, false>(smem + 8192, smem, wsP + P_WP1, wsP + P_SC1,
                            wsP + P_SH1, invBuf, pooled, base, N, l16, half);
  __syncthreads();
  run_layer<128, 256, false>(smem, smem + 8192, wsP + P_WP2, wsP + P_SC2,
                             wsP + P_SH2, invBuf, pooled, base, N, l16, half);
  __syncthreads();
  run_layer<256, 64, true>(smem + 8192, smem, wsP + P_WP3, wsP + P_SC3,
                           wsP + P_SH3, invBuf, pooled, base, N, l16, half);
}

// Decode encoded max back to float; rows >= M (no voxel) become 0.
__global__ void k_final(unsigned* pooled, const unsigned* __restrict__ Mp, int total) {
  int i = blockIdx.x * blockDim.x + threadIdx.x;
  if (i >= total) return;
  int row = i >> 6;
  unsigned M = Mp[0];
  float y = ((unsigned)row < M) ? decF(pooled[i]) : 0.f;
  pooled[i] = __float_as_uint(y);
}

// ---------------------------------------------------------------------------
extern "C" void kernel_launch(void* const* d_in, const int* in_sizes, int n_in,
                              void* d_out, int out_size, void* d_ws, size_t ws_size,
                              hipStream_t stream) {
  // setup_inputs order: point_feat, point_index, params{bn0..bn3, lin0..lin3}
  const float* feat = (const float*)d_in[0];
  const int* pidx = (const int*)d_in[1];
  const float* g0 = (const float*)d_in[2], *be0 = (const float*)d_in[3];
  const float* m0 = (const float*)d_in[4], *v0 = (const float*)d_in[5];
  const float* g1 = (const float*)d_in[6], *be1 = (const float*)d_in[7];
  const float* m1 = (const float*)d_in[8], *v1 = (const float*)d_in[9];
  const float* g2 = (const float*)d_in[10], *be2 = (const float*)d_in[11];
  const float* m2 = (const float*)d_in[12], *v2 = (const float*)d_in[13];
  const float* g3 = (const float*)d_in[14], *be3 = (const float*)d_in[15];
  const float* m3 = (const float*)d_in[16], *v3 = (const float*)d_in[17];
  const float* W0 = (const float*)d_in[18], *b0 = (const float*)d_in[19];
  const float* W1 = (const float*)d_in[20], *b1 = (const float*)d_in[21];
  const float* W2 = (const float*)d_in[22], *b2 = (const float*)d_in[23];
  const float* W3 = (const float*)d_in[24], *b3 = (const float*)d_in[25];

  int N = in_sizes[0] / 9;   // 400000
  int P = N / BATCH;         // points per batch (B=4 per reference)

  unsigned* ws = (unsigned*)d_ws;
  unsigned* flags = ws + OFF_FLAGS;
  unsigned* scan = ws + OFF_SCAN;
  unsigned* bsum = ws + OFF_BSUM;
  unsigned* Mp = ws + OFF_M;
  float* wsP = (float*)(ws + OFF_P);

  int* outIdx = (int*)d_out;                                      // [N,4] int32
  unsigned* pooled = (unsigned*)((float*)d_out + (size_t)N * 4);  // [N,64] f32

  int totPooled = N * 64;

  k_fill_u32<<<(NKEYS + 255) / 256, 256, 0, stream>>>(flags, 0u, NKEYS);
  k_fill_u32<<<(N * 4 + 255) / 256, 256, 0, stream>>>((unsigned*)outIdx, 0xFFFFFFFFu, N * 4);
  k_fill_u32<<<(totPooled + 255) / 256, 256, 0, stream>>>(pooled, ENC_NEG_INF, totPooled);

  k_param_prep<<<1, 256, 0, stream>>>(g0, be0, m0, v0, g1, be1, m1, v1,
                                      g2, be2, m2, v2, g3, be3, m3, v3,
                                      b0, b1, b2, b3, wsP);
  k_pack<<<(12 * 64 + 255) / 256, 256, 0, stream>>>(W0, wsP + P_WP0, 12, 64, 9);
  k_pack<<<(64 * 128 + 255) / 256, 256, 0, stream>>>(W1, wsP + P_WP1, 64, 128, 64);
  k_pack<<<(128 * 256 + 255) / 256, 256, 0, stream>>>(W2, wsP + P_WP2, 128, 256, 128);
  k_pack<<<(256 * 64 + 255) / 256, 256, 0, stream>>>(W3, wsP + P_WP3, 256, 64, 256);

  k_mark<<<(N + 255) / 256, 256, 0, stream>>>(pidx, flags, N, P);
  k_bsum<<<NKEYS / 1024, 256, 0, stream>>>(flags, bsum);
  k_bscan<<<1, 512, 0, stream>>>(bsum, Mp);
  k_scan<<<NKEYS / 1024, 256, 0, stream>>>(flags, bsum, scan);
  k_emit<<<(NKEYS + 255) / 256, 256, 0, stream>>>(flags, scan, outIdx);

  k_mlp<<<(N + 63) / 64, 32, 0, stream>>>(feat, pidx, scan, wsP, pooled, N, P);

  k_final<<<(totPooled + 255) / 256, 256, 0, stream>>>(pooled, Mp, totPooled);
}